// ROIAdaptiveEnsemble_134
// MI455X (gfx1250) — compile-verified
//
#include <hip/hip_runtime.h>

// out[b,t,v] = preds[idx[v], b, t, v]
// preds: (E=4, B=2, T=256, V=80000) f32, idx: (V,) i32 in [0,4), out: (2,256,80000) f32
//
// Memory-bound gather. Random idx at cacheline granularity forces reading ~all
// 4 epoch planes regardless of strategy, so we stream all 4 planes with fully
// coalesced non-temporal B128 loads and select in registers (v_cndmask), which
// keeps every VMEM request 100% lane-utilized. Roofline: ~819 MB @ 23.3 TB/s.

typedef float f32x4 __attribute__((ext_vector_type(4)));
typedef int   i32x4 __attribute__((ext_vector_type(4)));

namespace {
constexpr int      kB  = 2;
constexpr int      kT  = 256;
constexpr int      kV  = 80000;
constexpr int      kBT = kB * kT;                       // 512 rows
constexpr unsigned kPlane = (unsigned)kBT * (unsigned)kV; // 40,960,000 elems / epoch
constexpr int      kV4 = kV / 4;                        // 20000 float4 columns
constexpr int      kRowsPerThread = 8;                  // bt rows per thread
constexpr int      kBlock = 256;                        // 8 wave32s per block
}

__device__ __forceinline__ float sel4(int e, float a, float b, float c, float d) {
    float lo = (e & 1) ? b : a;   // v_cndmask
    float hi = (e & 1) ? d : c;   // v_cndmask
    return   (e & 2) ? hi : lo;   // v_cndmask
}

__global__ __launch_bounds__(kBlock) void epoch_select_kernel(
    const float* __restrict__ preds,
    const int*   __restrict__ vidx,
    float*       __restrict__ out)
{
    const int v4 = blockIdx.x * kBlock + threadIdx.x;
    if (v4 >= kV4) return;
    const unsigned v = (unsigned)v4 * 4u;

    // Per-voxel epoch indices: loaded once per thread, reused for all 8 rows.
    // Regular (temporal) load: shared by 64 grid.y stripes via L2.
    const i32x4 id = *(const i32x4*)(vidx + v);

    const unsigned bt0 = blockIdx.y * (unsigned)kRowsPerThread;

    const float* __restrict__ p0 = preds;
    const float* __restrict__ p1 = preds + (size_t)kPlane;
    const float* __restrict__ p2 = preds + 2 * (size_t)kPlane;
    const float* __restrict__ p3 = preds + 3 * (size_t)kPlane;

    // Prefetch the second half of our rows while the first half streams in.
    // Lowers to gfx1250 global_prefetch_b8 (speculative, NT).
#pragma unroll
    for (int k = kRowsPerThread / 2; k < kRowsPerThread; ++k) {
        const unsigned row = (bt0 + (unsigned)k) * (unsigned)kV + v;
        __builtin_prefetch(p0 + row, 0, 0);
        __builtin_prefetch(p1 + row, 0, 0);
        __builtin_prefetch(p2 + row, 0, 0);
        __builtin_prefetch(p3 + row, 0, 0);
    }

#pragma unroll 4
    for (int k = 0; k < kRowsPerThread; ++k) {
        const unsigned row = (bt0 + (unsigned)k) * (unsigned)kV + v;

        // Streaming reads: each preds byte is consumed exactly once -> TH_NT.
        const f32x4 a = __builtin_nontemporal_load((const f32x4*)(p0 + row));
        const f32x4 b = __builtin_nontemporal_load((const f32x4*)(p1 + row));
        const f32x4 c = __builtin_nontemporal_load((const f32x4*)(p2 + row));
        const f32x4 d = __builtin_nontemporal_load((const f32x4*)(p3 + row));

        f32x4 r;
        r.x = sel4(id.x, a.x, b.x, c.x, d.x);
        r.y = sel4(id.y, a.y, b.y, c.y, d.y);
        r.z = sel4(id.z, a.z, b.z, c.z, d.z);
        r.w = sel4(id.w, a.w, b.w, c.w, d.w);

        // Output written once, never re-read by this dispatch -> TH_NT store.
        __builtin_nontemporal_store(r, (f32x4*)(out + row));
    }
}

extern "C" void kernel_launch(void* const* d_in, const int* in_sizes, int n_in,
                              void* d_out, int out_size, void* d_ws, size_t ws_size,
                              hipStream_t stream) {
    const float* preds = (const float*)d_in[0];   // (4, 2, 256, 80000) f32
    const int*   vidx  = (const int*)d_in[1];     // (80000,) i32
    float*       out   = (float*)d_out;           // (2, 256, 80000) f32

    (void)in_sizes; (void)n_in; (void)out_size; (void)d_ws; (void)ws_size;

    dim3 grid((kV4 + kBlock - 1) / kBlock,        // 79 blocks over v columns
              kBT / kRowsPerThread);              // 64 stripes over bt rows
    epoch_select_kernel<<<grid, kBlock, 0, stream>>>(preds, vidx, out);
}